// BoundaryLoss_420906795782
// MI455X (gfx1250) — compile-verified
//
#include <hip/hip_runtime.h>
#include <hip/hip_bf16.h>
#include <cstdint>

#define NB 16
#define NH 384
#define NW 384
#define NPART 4096

typedef __attribute__((ext_vector_type(2))) float v2f;
typedef __attribute__((ext_vector_type(8))) float v8f;
typedef __attribute__((ext_vector_type(4))) unsigned int v4u;
typedef __attribute__((ext_vector_type(4))) int v4i;
typedef __attribute__((ext_vector_type(8))) int v8i;

// ---------------- Phase 0: zero partial-sum slots ----------------
__global__ void bl_zero_partials(float* __restrict__ part) {
    int id = blockIdx.x * blockDim.x + threadIdx.x;
    if (id < NPART) part[id] = 0.0f;
}

// ---------------- Phase 1: row-wise 1D distance, squared ----------------
// Exactly mirrors the reference cummax formulation (sentinel -BIG, clip BIG).
__global__ void bl_rowdist(const int* __restrict__ targets, float* __restrict__ g2) {
    int id = blockIdx.x * blockDim.x + threadIdx.x;
    if (id >= NB * NH) return;
    const int*  trow = targets + (size_t)id * NW;
    float*      grow = g2      + (size_t)id * NW;
    const float BIG = (float)(NH + NW);  // 768
    float run = -BIG;
    for (int j = 0; j < NW; ++j) {
        float v = (trow[j] > 0) ? (float)j : -BIG;
        run = fmaxf(run, v);
        grow[j] = (float)j - run;          // d_left
    }
    float run2 = -BIG;
    for (int j = NW - 1; j >= 0; --j) {
        float v = (trow[j] > 0) ? -(float)j : -BIG;
        run2 = fmaxf(run2, v);
        float dr = -run2 - (float)j;       // d_right
        float g  = fminf(fminf(grow[j], dr), BIG);
        grow[j]  = g * g;
    }
}

// ---------------- Phase 2: 2D EDT min-plus + loss, TDM-staged tiles ----------------
// grid = (NW/16, NH/64, NB), block = 256. Each block owns a 64-row x 16-col
// output tile; TDM loads g2[b][0..383][j0..j0+15] (16x384 tile, stride NW)
// into LDS, then each thread computes 4 pixels with a 384-deep fma/min loop.
__launch_bounds__(256)
__global__ void bl_dist_loss(const float* __restrict__ logits,
                             const int*   __restrict__ targets,
                             const float* __restrict__ g2,
                             float*       __restrict__ part) {
    __shared__ __align__(16) float g2s[NH * 16];   // [ip][jl], 24576 B
    __shared__ float red[256];

    const int tid = threadIdx.x;
    const int j0  = blockIdx.x * 16;
    const int i0  = blockIdx.y * 64;
    const int b   = blockIdx.z;

    // --- Tensor Data Mover: global -> LDS tile (issued by wave 0 only) ---
    if (tid < 32) {
        unsigned long long ga =
            (unsigned long long)(uintptr_t)(g2 + ((size_t)b * NH * NW + j0));
        unsigned int lds = (unsigned int)(uintptr_t)(&g2s[0]);
        v4u g0;
        g0[0] = 1u;                                   // count=1, user mode
        g0[1] = lds;                                  // lds_addr (bytes)
        g0[2] = (unsigned int)(ga & 0xFFFFFFFFu);     // global_addr[31:0]
        g0[3] = (unsigned int)(ga >> 32) | 0x80000000u; // ga[56:32] | type=2
        v8i g1;
        g1[0] = (int)(2u << 16);            // workgroup_mask=0, data_size=4B
        g1[1] = (int)((unsigned)NW << 16);  // tensor_dim0[15:0] -> bits[63:48]
        g1[2] = (int)((unsigned)NH << 16);  // tensor_dim1[15:0] -> bits[95:80]
        g1[3] = (int)(16u << 16);           // tile_dim0=16      -> bits[127:112]
        g1[4] = (int)NH;                    // tile_dim1=384     -> bits[143:128]
        g1[5] = (int)NW;                    // tensor_dim0_stride[31:0]
        g1[6] = 0;
        g1[7] = 0;
        v4i z4 = {0, 0, 0, 0};
        v8i z8 = {0, 0, 0, 0, 0, 0, 0, 0};
        __builtin_amdgcn_tensor_load_to_lds(g0, g1, z4, z4, z8, 0);
        __builtin_amdgcn_s_wait_tensorcnt(0);
    }
    __syncthreads();

    const int r  = tid >> 2;         // local row 0..63
    const int c4 = (tid & 3) << 2;   // local col base 0,4,8,12
    const int i  = i0 + r;

    float d0 = 3.0e30f, d1 = 3.0e30f, d2v = 3.0e30f, d3 = 3.0e30f;
    float di = (float)i;             // i - ip, starts at ip=0
#pragma unroll 4
    for (int ip = 0; ip < NH; ++ip) {
        float4 g4 = *(const float4*)&g2s[ip * 16 + c4];   // ds_load_b128
        float di2 = di * di;
        d0  = fminf(d0,  di2 + g4.x);
        d1  = fminf(d1,  di2 + g4.y);
        d2v = fminf(d2v, di2 + g4.z);
        d3  = fminf(d3,  di2 + g4.w);
        di -= 1.0f;
    }

    size_t pix = ((size_t)b * NH + i) * NW + j0 + c4;
    float4 x = *(const float4*)(logits + pix);
    int4   t = *(const int4*)(targets + pix);
    float p0 = 1.0f / (1.0f + expf(-x.x));
    float p1 = 1.0f / (1.0f + expf(-x.y));
    float p2 = 1.0f / (1.0f + expf(-x.z));
    float p3 = 1.0f / (1.0f + expf(-x.w));
    float acc = fabsf(p0 - (float)t.x) * sqrtf(d0)
              + fabsf(p1 - (float)t.y) * sqrtf(d1)
              + fabsf(p2 - (float)t.z) * sqrtf(d2v)
              + fabsf(p3 - (float)t.w) * sqrtf(d3);

    red[tid] = acc;
    __syncthreads();
    for (int s = 128; s > 0; s >>= 1) {
        if (tid < s) red[tid] += red[tid + s];
        __syncthreads();
    }
    if (tid == 0) {
        int wg = ((int)blockIdx.z * gridDim.y + blockIdx.y) * gridDim.x + blockIdx.x;
        part[wg] = red[0];
    }
}

// ---------------- Phase 3: WMMA-based final reduction (one wave) ----------------
// Sum 4096 partials via D = A x ones + C chained accumulation:
// A[m,k] = part[it*64 + k*16 + m]; row-sums accumulate, D replicated over N.
__launch_bounds__(32)
__global__ void bl_final_wmma(const float* __restrict__ part, float* __restrict__ out) {
    const int lane = threadIdx.x;
    v8f c = {0.f, 0.f, 0.f, 0.f, 0.f, 0.f, 0.f, 0.f};
    v2f one; one.x = 1.0f; one.y = 1.0f;
    // A 16x4 f32 layout: lanes 0-15 hold K=0 (v0), K=1 (v1); lanes 16-31 K=2,3.
    const int baseL = ((lane & 16) << 1) + (lane & 15);
    for (int it = 0; it < NPART / 64; ++it) {
        const int base = it * 64 + baseL;
        v2f a; a.x = part[base]; a.y = part[base + 16];
        c = __builtin_amdgcn_wmma_f32_16x16x4_f32(false, a, false, one,
                                                  (short)0, c, false, false);
    }
    // D column N=0: lane0 holds M=0..7 in c[0..7], lane16 holds M=8..15.
    float s = 0.0f;
#pragma unroll
    for (int k = 0; k < 8; ++k) s += c[k];
    s += __shfl_xor(s, 16, 32);
    if (lane == 0) out[0] = s / (float)((long long)NB * NH * NW);
}

extern "C" void kernel_launch(void* const* d_in, const int* in_sizes, int n_in,
                              void* d_out, int out_size, void* d_ws, size_t ws_size,
                              hipStream_t stream) {
    const float* logits  = (const float*)d_in[0];
    const int*   targets = (const int*)d_in[1];
    float* part = (float*)d_ws;
    float* g2   = (float*)((char*)d_ws + (size_t)NPART * sizeof(float));

    bl_zero_partials<<<NPART / 256, 256, 0, stream>>>(part);
    bl_rowdist<<<(NB * NH + 255) / 256, 256, 0, stream>>>(targets, g2);
    dim3 grid(NW / 16, NH / 64, NB);   // 24 x 6 x 16 = 2304 workgroups
    bl_dist_loss<<<grid, 256, 0, stream>>>(logits, targets, g2, part);
    bl_final_wmma<<<1, 32, 0, stream>>>(part, (float*)d_out);
}